// MeshConv_24678882083513
// MI455X (gfx1250) — compile-verified
//
#include <hip/hip_runtime.h>
#include <hip/hip_bf16.h>
#include <stddef.h>
#include <stdint.h>

// Problem constants (from the reference)
static constexpr int B_   = 8;
static constexpr int C_   = 64;     // C_IN == C_OUT == 64
static constexpr int V_   = 40962;
static constexpr int F_   = 81920;
static constexpr int NV_  = 40962;  // NV_PREV

typedef float v2f __attribute__((ext_vector_type(2)));
typedef float v8f __attribute__((ext_vector_type(8)));

__device__ __forceinline__ void fma4(float s, const float4& q, float4& a) {
    a.x = fmaf(s, q.x, a.x);
    a.y = fmaf(s, q.y, a.y);
    a.z = fmaf(s, q.z, a.z);
    a.w = fmaf(s, q.w, a.w);
}

// ---------------------------------------------------------------------------
// Kernel 0: transpose x[b][c][v] -> xT[b][v][c] so every downstream gather
// (by vertex index) reads a 256-B channel-contiguous row. Classic LDS tile
// transpose, 64x64 tiles, both phases coalesced.
// ---------------------------------------------------------------------------
__global__ void __launch_bounds__(256) mesh_transpose_x(
    const float* __restrict__ x,   // [B][C][V]
    float* __restrict__ xT)        // [B][V][C]
{
    __shared__ float tile[64][65];
    const int tid = threadIdx.x;
    const int b   = blockIdx.y;
    const int v0  = blockIdx.x * 64;
    const int col = tid & 63;
    const int r0  = tid >> 6;      // 0..3

#pragma unroll
    for (int i = 0; i < 16; ++i) {
        const int r = r0 + 4 * i;              // channel index
        const int v = v0 + col;
        tile[r][col] = (v < V_) ? x[((size_t)b * C_ + r) * V_ + v] : 0.0f;
    }
    __syncthreads();
#pragma unroll
    for (int i = 0; i < 16; ++i) {
        const int r = r0 + 4 * i;              // vertex offset in tile
        const int v = v0 + r;
        if (v < V_)
            xT[((size_t)b * V_ + v) * C_ + col] = tile[col][r];
    }
}

// ---------------------------------------------------------------------------
// Kernel 1: per-face gradient dotted with EW/NS frame vectors.
//   grad_face[k] = sum_{j<3} G_val[3*(k*F+f)+j] * xT[b][G_col][c]
//   gfe[b][f][c] = sum_k grad_face[k]*EW[f][k];  gfn likewise with NS.
// Block: 256 threads = 4 faces x 64 channels(lane). With xT layout every
// gather is a coalesced 256-B row read; writes are lane-contiguous.
// ---------------------------------------------------------------------------
__global__ void __launch_bounds__(256) mesh_face_grad(
    const float* __restrict__ xT,    // [B][V][C]
    const int*   __restrict__ Gcol,  // [3F*3]
    const float* __restrict__ Gval,  // [3F*3]
    const float* __restrict__ EW,    // [F][3]
    const float* __restrict__ NS,    // [F][3]
    float* __restrict__ gfe,         // [B][F][C]
    float* __restrict__ gfn)         // [B][F][C]
{
    const int c    = threadIdx.x & 63;
    const int fsub = threadIdx.x >> 6;
    const int f    = blockIdx.x * 4 + fsub;
    const int b    = blockIdx.y;
    if (f >= F_) return;

    const float* xb = xT + (size_t)b * V_ * C_ + c;

    float s[3];
#pragma unroll
    for (int k = 0; k < 3; ++k) {
        const size_t base = 3 * ((size_t)k * F_ + f);
        float acc = 0.0f;
#pragma unroll
        for (int j = 0; j < 3; ++j) {
            const int col = Gcol[base + j];
            acc = fmaf(Gval[base + j], xb[(size_t)col * C_], acc);
        }
        s[k] = acc;
    }

    const float ew = fmaf(s[0], EW[3 * f + 0],
                     fmaf(s[1], EW[3 * f + 1], s[2] * EW[3 * f + 2]));
    const float ns = fmaf(s[0], NS[3 * f + 0],
                     fmaf(s[1], NS[3 * f + 1], s[2] * NS[3 * f + 2]));

    const size_t o = ((size_t)b * F_ + f) * (size_t)C_ + c;
    gfe[o] = ew;
    gfn[o] = ns;
}

// ---------------------------------------------------------------------------
// Kernel 2: fused vertex gather + einsum('bcvk,ock->bov') via fp32 WMMA.
//
// GEMM view: out[b][o][v] = sum_{ck} coeffs[o][ck] * res[ck][v],
//   ck = c*4 + kcoef, M = 64 (o), K = 256, N = vertices.
// fp32 16x16x4 B-fragment layout: lane(n,hi): VGPR0=B[2hi][n], VGPR1=B[2hi+1][n]
//   -> hi=0 lanes hold (identity, grad_ew); hi=1 lanes hold (grad_ns, lap).
// Branchless: hi=0 lanes carry a 1-nnz "identity row" {col=v,val=1} in the
// same 7-term x-dot hi=1 lanes use for the laplacian; v_cndmask routes the
// (grad, xdot) pair into the fragment. EXEC stays all-ones throughout.
// All gathers are c-contiguous 256-B rows -> float4 loads, 13 b128 / 4 chans.
// Coeffs staged in LDS via global_load_async_to_lds_b128 (ASYNCcnt path).
// ---------------------------------------------------------------------------
__global__ void __launch_bounds__(256, 1) mesh_conv_gemm(
    const float* __restrict__ xT,      // [B][V][C]
    const float* __restrict__ gfe,     // [B][F][C]
    const float* __restrict__ gfn,     // [B][F][C]
    const int*   __restrict__ F2Vcol,  // [NV*6]
    const float* __restrict__ F2Vval,  // [NV*6]
    const int*   __restrict__ Lcol,    // [NV*7]
    const float* __restrict__ Lval,    // [NV*7]
    const float* __restrict__ coeffs,  // [64][64][4]
    float* __restrict__ out)           // [B][64][NV]
{
    __shared__ float Alds[64 * 256];   // 64 KB: coeffs[o][ck]

    const int tid = threadIdx.x;
    {
        // Async copy 64 KB of coeffs into LDS: 4096 b128, 16 per thread.
        const uint64_t gbase = (uint64_t)(uintptr_t)coeffs + (uint64_t)tid * 16u;
        const uint32_t lbase = (uint32_t)(uintptr_t)(&Alds[0]) + (uint32_t)tid * 16u;
#pragma unroll
        for (int i = 0; i < 16; ++i) {
            const uint64_t ga = gbase + (uint64_t)i * 256u * 16u;
            const uint32_t la = lbase + (uint32_t)i * 256u * 16u;
            asm volatile("global_load_async_to_lds_b128 %0, %1, off"
                         :: "v"(la), "v"(ga) : "memory");
        }
        asm volatile("s_wait_asynccnt 0x0" ::: "memory");
    }
    __syncthreads();

    const int b    = blockIdx.y;
    const int lane = tid & 31;
    const int wave = tid >> 5;                  // 0..7
    const int n    = lane & 15;
    const int hi   = lane >> 4;
    const int v    = blockIdx.x * 128 + wave * 16 + n;
    const int vc   = (v < NV_) ? v : (NV_ - 1); // clamp: keep EXEC all-ones

    const float* xb   = xT + (size_t)b * V_ * C_;
    const float* gsrc = (hi ? gfn : gfe) + (size_t)b * F_ * C_;

    // 6-face gather rows (gfe for hi=0 lanes, gfn for hi=1 lanes).
    const float* gptr[6];
    float        fval[6];
#pragma unroll
    for (int j = 0; j < 6; ++j) {
        const int fc = F2Vcol[6 * vc + j];
        fval[j] = F2Vval[6 * vc + j];
        gptr[j] = gsrc + (size_t)fc * C_;
    }
    // 7-term x-dot rows: identity (hi=0) or laplacian (hi=1), lane-dependent.
    const float* xrow[7];
    float        lval[7];
#pragma unroll
    for (int j = 0; j < 7; ++j) {
        int   lc, ok = (hi != 0);
        lc      = ok ? Lcol[7 * vc + j] : vc;
        lval[j] = ok ? Lval[7 * vc + j] : (j == 0 ? 1.0f : 0.0f);
        xrow[j] = xb + (size_t)lc * C_;
    }

    // Warm caches (global_prefetch_b8).
#pragma unroll
    for (int j = 0; j < 6; ++j) __builtin_prefetch(gptr[j], 0, 1);
#pragma unroll
    for (int j = 0; j < 7; ++j) __builtin_prefetch(xrow[j], 0, 1);

    v8f acc0 = {}, acc1 = {}, acc2 = {}, acc3 = {};

    for (int c4 = 0; c4 < 64; c4 += 4) {
        // Vectorized gathers: 13 x global_load_b128 per 4 channels.
        float4 bg = {0.f, 0.f, 0.f, 0.f};   // grad (ew or ns per half-wave)
#pragma unroll
        for (int j = 0; j < 6; ++j)
            fma4(fval[j], *(const float4*)(gptr[j] + c4), bg);

        float4 bx = {0.f, 0.f, 0.f, 0.f};   // identity (hi=0) / laplacian (hi=1)
#pragma unroll
        for (int j = 0; j < 7; ++j)
            fma4(lval[j], *(const float4*)(xrow[j] + c4), bx);

        const float gq[4] = {bg.x, bg.y, bg.z, bg.w};
        const float xq[4] = {bx.x, bx.y, bx.z, bx.w};

#pragma unroll
        for (int u = 0; u < 4; ++u) {
            const int c = c4 + u;
            v2f bfrag;
            bfrag.x = hi ? gq[u] : xq[u];   // K=2hi+0 : identity | ns
            bfrag.y = hi ? xq[u] : gq[u];   // K=2hi+1 : ew | laplacian

            // A fragments: row o = 16*mt + n, K pair (c*4+2hi, +1), 8B aligned.
            const int akoff = (c << 2) + (hi << 1);
            const v2f a0 = *(const v2f*)&Alds[(0  + n) * 256 + akoff];
            const v2f a1 = *(const v2f*)&Alds[(16 + n) * 256 + akoff];
            const v2f a2 = *(const v2f*)&Alds[(32 + n) * 256 + akoff];
            const v2f a3 = *(const v2f*)&Alds[(48 + n) * 256 + akoff];

            acc0 = __builtin_amdgcn_wmma_f32_16x16x4_f32(false, a0, false, bfrag,
                                                         (short)0, acc0, false, false);
            acc1 = __builtin_amdgcn_wmma_f32_16x16x4_f32(false, a1, false, bfrag,
                                                         (short)0, acc1, false, false);
            acc2 = __builtin_amdgcn_wmma_f32_16x16x4_f32(false, a2, false, bfrag,
                                                         (short)0, acc2, false, false);
            acc3 = __builtin_amdgcn_wmma_f32_16x16x4_f32(false, a3, false, bfrag,
                                                         (short)0, acc3, false, false);
        }
    }

    // D layout: VGPR i -> (o = 16*mt + i + 8*hi, vertex = v). Mask tail only.
    if (v < NV_) {
        float* ob = out + (size_t)b * 64 * NV_ + v;
#pragma unroll
        for (int i = 0; i < 8; ++i) {
            const int o0 = i + 8 * hi;
            ob[(size_t)(o0 +  0) * NV_] = acc0[i];
            ob[(size_t)(o0 + 16) * NV_] = acc1[i];
            ob[(size_t)(o0 + 32) * NV_] = acc2[i];
            ob[(size_t)(o0 + 48) * NV_] = acc3[i];
        }
    }
}

// ---------------------------------------------------------------------------
// Host launcher. Input order (setup_inputs dict):
//  0:input 1:G_row 2:G_col 3:G_val 4:L_row 5:L_col 6:L_val
//  7:F2V_row 8:F2V_col 9:F2V_val 10:EW 11:NS 12:coeffs
// Row arrays are unused: rows == repeat(arange(nrows), nnz) by construction,
// so each spmatmul is fixed-width CSR with implicit row pointers.
// Workspace: xT (80 MiB) + gfe + gfn (160 MiB each).
// ---------------------------------------------------------------------------
extern "C" void kernel_launch(void* const* d_in, const int* in_sizes, int n_in,
                              void* d_out, int out_size, void* d_ws, size_t ws_size,
                              hipStream_t stream) {
    (void)in_sizes; (void)n_in; (void)out_size; (void)ws_size;

    const float* x      = (const float*)d_in[0];
    const int*   Gcol   = (const int*)  d_in[2];
    const float* Gval   = (const float*)d_in[3];
    const int*   Lcol   = (const int*)  d_in[5];
    const float* Lval   = (const float*)d_in[6];
    const int*   F2Vcol = (const int*)  d_in[8];
    const float* F2Vval = (const float*)d_in[9];
    const float* EW     = (const float*)d_in[10];
    const float* NS     = (const float*)d_in[11];
    const float* coeffs = (const float*)d_in[12];
    float*       out    = (float*)d_out;

    float* xT  = (float*)d_ws;
    float* gfe = xT + (size_t)B_ * V_ * C_;
    float* gfn = gfe + (size_t)B_ * F_ * C_;

    dim3 grid0((V_ + 63) / 64, B_);
    hipLaunchKernelGGL(mesh_transpose_x, grid0, dim3(256), 0, stream, x, xT);

    dim3 grid1(F_ / 4, B_);
    hipLaunchKernelGGL(mesh_face_grad, grid1, dim3(256), 0, stream,
                       xT, Gcol, Gval, EW, NS, gfe, gfn);

    dim3 grid2((NV_ + 127) / 128, B_);
    hipLaunchKernelGGL(mesh_conv_gemm, grid2, dim3(256), 0, stream,
                       xT, gfe, gfn, F2Vcol, F2Vval, Lcol, Lval, coeffs, out);
}